// DLPLoss_64656437674024
// MI455X (gfx1250) — compile-verified
//
#include <hip/hip_runtime.h>
#include <math.h>

typedef float v2f __attribute__((ext_vector_type(2)));
typedef float v8f __attribute__((ext_vector_type(8)));

#define HW     1024
#define NLINES 64

// d_ws float layout:
//   [0..63]        cs   = c_i  * (1/len_i)
//   [64..127]      dxs  = dx_i * (1/len_i)
//   [128..191]     dys  = dy_i * (1/len_i)
//   [192..255]     lens = len_i
//   [256..1279]    per-block dp partials (1024 blocks)
//   [1280..2303]   per-block dn partials

__global__ void dlp_prep(const float* __restrict__ gt_lines, float* __restrict__ ws) {
    int i = threadIdx.x;
    if (i < NLINES) {
        float p1x = gt_lines[i * 4 + 0];
        float p1y = gt_lines[i * 4 + 1];
        float p2x = gt_lines[i * 4 + 2];
        float p2y = gt_lines[i * 4 + 3];
        float dx = p2x - p1x;          // dvec[:,0]
        float dy = p2y - p1y;          // dvec[:,1]
        float len = sqrtf(dx * dx + dy * dy);
        float il = 1.0f / len;
        float c = dx * p1y - dy * p1x;
        ws[0   + i] = c  * il;
        ws[64  + i] = dx * il;
        ws[128 + i] = dy * il;
        ws[192 + i] = len;
    }
}

// Each wave: 8 tiles of 16 pixels; per tile, 4x V_WMMA_F32_16X16X4_F32 produce
// signed distances to all 64 lines (lines on M, pixels on N, coeffs pre-scaled
// by 1/len). Branch-free two-pass min/argmax selection, half-merge via shuffle.
__global__ __launch_bounds__(256) void dlp_main(const float* __restrict__ y_pred,
                                                const float* __restrict__ gtll,
                                                float* __restrict__ ws) {
    __shared__ float slens[NLINES];
    __shared__ float sdp[8], sdn[8];

    const int lane   = threadIdx.x & 31;
    const int wv     = threadIdx.x >> 5;
    const int gw     = blockIdx.x * 8 + wv;
    const int lo     = lane & 15;
    const bool lower = lane < 16;
    const int half_off = lower ? 0 : 8;

    const float* cs   = ws;
    const float* dxs  = ws + 64;
    const float* dys  = ws + 128;
    const float* lens = ws + 192;

    if (threadIdx.x < NLINES) slens[threadIdx.x] = lens[threadIdx.x];
    __syncthreads();

    // A fragments (16 lines x 4), row m = [c'_m, dx'_m, dy'_m, 0]
    // A layout: lanes 0-15 -> {K0,K1}, lanes 16-31 -> {K2,K3}, M = lane%16
    v2f A[4];
#pragma unroll
    for (int t = 0; t < 4; ++t) {
        int n = t * 16 + lo;
        A[t].x = lower ? cs[n]  : dys[n];
        A[t].y = lower ? dxs[n] : 0.0f;
    }

    float accdp = 0.0f, accdn = 0.0f;

    for (int j = 0; j < 8; ++j) {
        const int tilebase = (gw * 8 + j) * 16;
        const int p  = tilebase + lo;               // this lane's pixel
        const float xf = (float)(p & (HW - 1));
        const float yf = (float)(p >> 10);

        // B fragment (4 x 16 pixels), col n = [1, -x, y, 0]^T
        // B layout: V0 = {K0 low lanes, K2 high}, V1 = {K1 low, K3 high}
        v2f B;
        B.x = lower ? 1.0f : yf;
        B.y = lower ? -xf  : 0.0f;

        v8f D[4];
#pragma unroll
        for (int t = 0; t < 4; ++t) {
            v8f c0 = {0.0f, 0.0f, 0.0f, 0.0f, 0.0f, 0.0f, 0.0f, 0.0f};
            D[t] = __builtin_amdgcn_wmma_f32_16x16x4_f32(
                false, A[t], false, B, (short)0, c0, false, false);
        }

        // ---- Pass 1: masked min over candidate distances (branch-free) ----
        // Candidate: |d| <= 1 and global line index > 0 (only element t=0,r=0
        // in the lower half is line 0). Non-candidates -> +inf.
        const float INF = __builtin_inff();
        float m  = INF;
        float d0 = 0.0f;   // line-0 distance (valid in lower half)
#pragma unroll
        for (int t = 0; t < 4; ++t) {
#pragma unroll
            for (int r = 0; r < 8; ++r) {
                float dabs = fabsf(D[t][r]);
                int ok = (int)(dabs <= 1.0f);
                if (t == 0 && r == 0) {
                    d0 = dabs;
                    ok &= (int)(!lower);   // bitwise: no short-circuit branch
                }
                float de = ok ? dabs : INF;
                D[t][r] = de;              // keep masked value for pass 2
                m = fminf(m, de);
            }
        }
        // merge the two lane halves (lines split 0-7 / 8-15 per tile)
        m = fminf(m, __shfl_xor(m, 16, 32));

        // ---- Pass 2: last (max) index achieving the min ----
        int bi = -1;
#pragma unroll
        for (int t = 0; t < 4; ++t) {
#pragma unroll
            for (int r = 0; r < 8; ++r) {
                int idx = t * 16 + r + half_off;
                int ci  = (D[t][r] == m) ? idx : -1;
                bi = (ci > bi) ? ci : bi;
            }
        }
        {
            int ob = __shfl_xor(bi, 16, 32);
            bi = (ob > bi) ? ob : bi;
        }
        // broadcast d0 from the lower-half lane that owns this pixel
        d0 = __shfl(d0, lo, 32);

        int   sel = (m <= d0) ? bi : 0;    // m==inf -> fallback to line 0
        float ll  = slens[sel];

        float g  = gtll[p];
        float e  = g - ll;
        float e2 = e * e;
        bool  pos = (y_pred[p] != 0.0f);
        accdp += (lower &&  pos) ? e2 : 0.0f;
        accdn += (lower && !pos) ? e2 : 0.0f;
    }

    // deterministic wave reduction (upper-half contributions are zero)
#pragma unroll
    for (int msk = 16; msk >= 1; msk >>= 1) {
        accdp += __shfl_xor(accdp, msk, 32);
        accdn += __shfl_xor(accdn, msk, 32);
    }
    if (lane == 0) { sdp[wv] = accdp; sdn[wv] = accdn; }
    __syncthreads();
    if (threadIdx.x == 0) {
        float dp = 0.0f, dn = 0.0f;
#pragma unroll
        for (int w = 0; w < 8; ++w) { dp += sdp[w]; dn += sdn[w]; }
        ws[256 + blockIdx.x]        = dp;
        ws[256 + 1024 + blockIdx.x] = dn;
    }
}

__global__ __launch_bounds__(256) void dlp_final(const float* __restrict__ ws,
                                                 float* __restrict__ out) {
    __shared__ float sdp[256], sdn[256];
    int t = threadIdx.x;
    float dp = 0.0f, dn = 0.0f;
    for (int k = t; k < 1024; k += 256) {
        dp += ws[256 + k];
        dn += ws[256 + 1024 + k];
    }
    sdp[t] = dp; sdn[t] = dn;
    __syncthreads();
    for (int s = 128; s > 0; s >>= 1) {
        if (t < s) { sdp[t] += sdp[t + s]; sdn[t] += sdn[t + s]; }
        __syncthreads();
    }
    if (t == 0) {
        float DP = sdp[0], DN = sdn[0];
        float tot = DP + DN;
        out[0] = DN / tot * DN + DP / tot * DP;
    }
}

extern "C" void kernel_launch(void* const* d_in, const int* in_sizes, int n_in,
                              void* d_out, int out_size, void* d_ws, size_t ws_size,
                              hipStream_t stream) {
    const float* y_pred   = (const float*)d_in[0];   // (1024,1024) f32
    const float* gtll     = (const float*)d_in[1];   // (1024,1024) f32
    const float* gt_lines = (const float*)d_in[2];   // (64,2,2)    f32
    float* out = (float*)d_out;
    float* ws  = (float*)d_ws;

    dlp_prep<<<1, 64, 0, stream>>>(gt_lines, ws);
    dlp_main<<<1024, 256, 0, stream>>>(y_pred, gtll, ws);
    dlp_final<<<1, 256, 0, stream>>>(ws, out);
}